// CustomKernelConvLatticeIm2RowModule_39384850105055
// MI455X (gfx1250) — compile-verified
//
#include <hip/hip_runtime.h>

typedef float v2f __attribute__((ext_vector_type(2)));
typedef float v8f __attribute__((ext_vector_type(8)));

#define FDIM 64
#define KN 9

// One wave32 per vertex. Neighbor rows are gathered directly in the
// V_WMMA_F32_16X16X4_F32 B-operand layout; each lane's distance/weight work
// is assigned so its results ARE the WMMA A-operand components. Since all A
// rows are identical, D rows are identical too, so both wave halves carry
// valid aflow values -> fully unpredicated full-wave stores.
__global__ __launch_bounds__(256) void aflow_wmma_kernel(
    const float* __restrict__ lattice,   // N x 64
    const float* __restrict__ hidden,    // N x 64
    const float* __restrict__ bias,      // 64
    const float* __restrict__ alpha_p,   // 1
    const float* __restrict__ beta_p,    // 1
    const int*   __restrict__ nidx,      // N x 9
    float* __restrict__ out_aflow,       // N x 64
    float* __restrict__ out_w,           // N x 9
    float* __restrict__ out_idx,         // N x 9 (numeric float of int idx)
    int N)
{
    const int lane = threadIdx.x & 31;
    const int wave = threadIdx.x >> 5;
    const int v    = blockIdx.x * 8 + wave;
    if (v >= N) return;                      // whole-wave uniform guard

    const float alpha = alpha_p[0];
    const float beta  = beta_p[0];
    const bool  hi    = (lane >= 16);
    const int   col   = lane & 15;

    // full-wave output column split: lanes 0-15 -> f0=col,    f1=col+16
    //                                lanes 16-31 -> f0=col+32, f1=col+48
    const int f0 = (hi ? 32 : 0) + col;
    const int f1 = f0 + 16;

    // ---- preload bias early (overlaps gather latency) ----
    const float bias0 = bias[f0];
    const float bias1 = bias[f1];

    // ---- neighbor indices (wave-uniform, constant-indexed only) ----
    int idxv[12];
    #pragma unroll
    for (int k = 0; k < KN; ++k) idxv[k] = nidx[(long)v * KN + k];
    #pragma unroll
    for (int k = KN; k < 12; ++k) idxv[k] = -1;   // padding rows invalid

    // ---- center row, 4 column tiles of 16 (both halves read same 64B) ----
    float c[4];
    #pragma unroll
    for (int t = 0; t < 4; ++t)
        c[t] = lattice[(long)v * FDIM + 16 * t + col];

    // Per-chunk per-lane row assignment:
    //   lanes 0-15  handle rows 4kc+0 (.x), 4kc+1 (.y)
    //   lanes 16-31 handle rows 4kc+2 (.x), 4kc+3 (.y)
    v2f   b[3][4];         // WMMA B operands (constant-indexed)
    float sqx[3], sqy[3];  // this lane's reduced squared distances
    int   vx[3],  vy[3];   // this lane's validity flags
    float halfsum = 0.0f;  // sum of distances over this half's rows

    #pragma unroll
    for (int kc = 0; kc < 3; ++kc) {
        const int iA0 = idxv[4 * kc + 0];
        const int iA1 = idxv[4 * kc + 1];
        const int iB0 = idxv[4 * kc + 2];
        const int iB1 = idxv[4 * kc + 3];
        const int i0  = hi ? iB0 : iA0;      // value-selects, single cndmask
        const int i1  = hi ? iB1 : iA1;
        const int v0  = (i0 >= 0);
        const int v1  = (i1 >= 0);
        vx[kc] = v0;
        vy[kc] = v1;

        const float* p0 = hidden + (long)(v0 ? i0 : 0) * FDIM;
        const float* p1 = hidden + (long)(v1 ? i1 : 0) * FDIM;

        float s0 = 0.0f, s1 = 0.0f;
        #pragma unroll
        for (int t = 0; t < 4; ++t) {
            const int f = 16 * t + col;
            float x = v0 ? p0[f] : 0.0f;     // coalesced 64B per half-wave
            float y = v1 ? p1[f] : 0.0f;
            b[kc][t].x = x;
            b[kc][t].y = y;
            float d0 = x - c[t];
            float d1 = y - c[t];
            s0 = fmaf(d0, d0, s0);
            s1 = fmaf(d1, d1, s1);
        }
        // reduce over the 16 lanes of this half -> half-uniform sums
        #pragma unroll
        for (int m = 1; m <= 8; m <<= 1) {
            s0 += __shfl_xor(s0, m, 32);
            s1 += __shfl_xor(s1, m, 32);
        }
        sqx[kc] = s0;
        sqy[kc] = s1;
        float d0 = v0 ? __builtin_sqrtf(s0) : 0.0f;
        float d1 = v1 ? __builtin_sqrtf(s1) : 0.0f;
        halfsum += d0 + d1;
    }

    // total distance sum across both halves -> every lane
    const float dsum = halfsum + __shfl_xor(halfsum, 16, 32);
    const float inv  = 1.0f / dsum;          // one precise divide

    // ---- AFLOW weights: each lane's values are its WMMA A components ----
    float wxa[3], wya[3];
    #pragma unroll
    for (int kc = 0; kc < 3; ++kc) {
        float d0 = vx[kc] ? __builtin_sqrtf(sqx[kc]) : 0.0f;
        float d1 = vy[kc] ? __builtin_sqrtf(sqy[kc]) : 0.0f;
        float n0 = d0 * inv;
        float n1 = d1 * inv;
        float w0 = (alpha - fminf(n0, alpha)) * beta;
        float w1 = (alpha - fminf(n1, alpha)) * beta;
        wxa[kc] = vx[kc] ? w0 : 0.0f;
        wya[kc] = vy[kc] ? w1 : 0.0f;
    }

    // ---- aflow = W(1x12) * Neigh(12x64) on the f32 WMMA pipe ----
    float outv[4];
    #pragma unroll
    for (int t = 0; t < 4; ++t) {
        v8f acc = {};
        #pragma unroll
        for (int kc = 0; kc < 3; ++kc) {
            v2f a;
            a.x = wxa[kc];
            a.y = wya[kc];
            acc = __builtin_amdgcn_wmma_f32_16x16x4_f32(
                /*neg_a=*/false, a, /*neg_b=*/false, b[kc][t],
                /*c_mod=*/(short)0, acc, /*reuse_a=*/false, /*reuse_b=*/false);
        }
        // all D rows identical: VGPR0 of both halves holds aflow[16t + col]
        outv[t] = acc[0];
    }

    // full-wave stores: low half stores tiles 0,1; high half stores tiles 2,3
    {
        const float o0 = outv[0], o1 = outv[1], o2 = outv[2], o3 = outv[3];
        const float s0 = (hi ? o2 : o0) + bias0;
        const float s1 = (hi ? o3 : o1) + bias1;
        out_aflow[(long)v * FDIM + f0] = s0;
        out_aflow[(long)v * FDIM + f1] = s1;
    }

    // ---- weights output: one predicated full-wave scatter ----
    // designated lanes: low half lanes {0,1,4,5,8} (slots 0,1),
    //                   high half lanes {18,19,22,23} (slots 2,3)
    {
        const int   slot = lane & 3;
        const int   kcl  = (lane & 15) >> 2;
        const bool  mask = ((lane & 15) < KN) && ((slot < 2) == !hi);
        // constant-index extraction, then value-only selects
        const float x0 = wxa[0], x1 = wxa[1], x2 = wxa[2];
        const float y0 = wya[0], y1 = wya[1], y2 = wya[2];
        float px = x0; px = (kcl == 1) ? x1 : px; px = (kcl == 2) ? x2 : px;
        float py = y0; py = (kcl == 1) ? y1 : py; py = (kcl == 2) ? y2 : py;
        const float wval = (lane & 1) ? py : px;
        if (mask)
            out_w[(long)v * KN + (lane & 15)] = wval;
    }

    // ---- idx output: re-load (L2-hot) and convert ----
    if (lane < KN)
        out_idx[(long)v * KN + lane] = (float)nidx[(long)v * KN + lane];
}

extern "C" void kernel_launch(void* const* d_in, const int* in_sizes, int n_in,
                              void* d_out, int out_size, void* d_ws, size_t ws_size,
                              hipStream_t stream) {
    const float* lattice = (const float*)d_in[0];
    const float* hidden  = (const float*)d_in[1];
    const float* bias    = (const float*)d_in[2];
    const float* alpha   = (const float*)d_in[3];
    const float* beta    = (const float*)d_in[4];
    const int*   nidx    = (const int*)d_in[5];

    const int N = in_sizes[0] / FDIM;   // 300000

    float* out_aflow = (float*)d_out;                        // N*64
    float* out_w     = out_aflow + (long)N * FDIM;           // N*9
    float* out_idx   = out_w + (long)N * KN;                 // N*9

    const int wavesPerBlock = 8;                             // 256 threads, wave32
    const int grid = (N + wavesPerBlock - 1) / wavesPerBlock;
    aflow_wmma_kernel<<<grid, 256, 0, stream>>>(
        lattice, hidden, bias, alpha, beta, nidx,
        out_aflow, out_w, out_idx, N);
}